// QMixer_41472204210984
// MI455X (gfx1250) — compile-verified
//
#include <hip/hip_runtime.h>
#include <hip/hip_bf16.h>
#include <math.h>

#define BB    64
#define LL    1024
#define AA    20
#define FEA   512
#define SD    (AA * FEA)      // 10240
#define EMB   64
#define HYP   512

typedef __attribute__((ext_vector_type(2))) float v2f;
typedef __attribute__((ext_vector_type(8))) float v8f;

// ---------------------------------------------------------------------------
// Kernel 1: q_sele gather + segment sums (agent_qs, counts) per batch.
// One block per b; LDS float atomics (ds_add_f32) for the 20 bins.
// qs_wt is read once and q_sele written once -> non-temporal.
// ---------------------------------------------------------------------------
__global__ void k_select(const float* __restrict__ qs_wt,
                         const int* __restrict__ aa_wt,
                         const int* __restrict__ aa_mut,
                         const unsigned char* __restrict__ agent_mask,
                         const unsigned char* __restrict__ mask,
                         const unsigned char* __restrict__ mut_mask,
                         float* __restrict__ q_sele_out,   // [B,L]
                         float* __restrict__ agent_qs,     // [B,A]
                         float* __restrict__ counts) {     // [B,A]
    const int b   = blockIdx.x;
    const int tid = threadIdx.x;
    __shared__ float s_q[AA];
    __shared__ float s_c[AA];
    if (tid < AA) { s_q[tid] = 0.f; s_c[tid] = 0.f; }
    __syncthreads();
    for (int l = tid; l < LL; l += blockDim.x) {
        const int idx    = b * LL + l;
        const bool valid = agent_mask[idx] && mask[idx];
        const int aw  = aa_wt[idx];
        const int am  = aa_mut[idx];
        const int sel = mut_mask[idx] ? am : aw;
        const float q = valid
            ? __builtin_nontemporal_load(qs_wt + (long)idx * AA + sel) : 0.f;
        __builtin_nontemporal_store(q, q_sele_out + idx);
        if (valid) {
            atomicAdd(&s_q[aw], q);
            atomicAdd(&s_c[aw], 1.f);
            atomicAdd(&s_c[am], 1.f);
        }
    }
    __syncthreads();
    if (tid < AA) {
        agent_qs[b * AA + tid] = s_q[tid];
        counts  [b * AA + tid] = s_c[tid];
    }
}

// ---------------------------------------------------------------------------
// Kernel 2: segment-mean of embeddings -> states[b, a*512 + f].
// Grid = B*4 blocks of 128 threads; each block owns (b, 128-wide f chunk).
// Each thread owns ONE f column -> conflict-free LDS accumulation.
// Invalid rows skipped (block-uniform branch) -> ~half the HBM traffic.
// Manual unroll-by-4: load phase (<=8 independent NT global loads, clause-
// able) then accumulate phase, in fixed l order (deterministic fp sums).
// Embeddings are streamed once -> non-temporal so the ~50 MB of hypernet
// weights stay resident in the 192 MB L2 for the GEMMs that follow.
// ---------------------------------------------------------------------------
__global__ void k_gstate(const int* __restrict__ aa_wt,
                         const int* __restrict__ aa_mut,
                         const unsigned char* __restrict__ agent_mask,
                         const unsigned char* __restrict__ mask,
                         const float* __restrict__ emb_wt,
                         const float* __restrict__ emb_mut,
                         const float* __restrict__ counts,
                         float* __restrict__ states) {
    const int b   = blockIdx.x >> 2;
    const int f0  = (blockIdx.x & 3) * 128;
    const int tid = threadIdx.x;           // 0..127
    __shared__ int   codes[LL];            // 4 KB
    __shared__ float acc[AA * 128];        // 10 KB

    for (int i = tid; i < AA * 128; i += 128) acc[i] = 0.f;
    for (int l = tid; l < LL; l += 128) {
        const int idx = b * LL + l;
        const bool valid = agent_mask[idx] && mask[idx];
        codes[l] = valid ? (aa_wt[idx] | (aa_mut[idx] << 8)) : -1;
    }
    __syncthreads();

    const float* ew = emb_wt  + (long)b * LL * FEA + f0 + tid;
    const float* em = emb_mut + (long)b * LL * FEA + f0 + tid;
    for (int l = 0; l < LL; l += 4) {
        const int c0 = codes[l + 0];
        const int c1 = codes[l + 1];
        const int c2 = codes[l + 2];
        const int c3 = codes[l + 3];
        float w0 = 0.f, w1 = 0.f, w2 = 0.f, w3 = 0.f;
        float u0 = 0.f, u1 = 0.f, u2 = 0.f, u3 = 0.f;
        // ---- load phase (independent, non-temporal) ----
        if (c0 >= 0) { w0 = __builtin_nontemporal_load(ew + (long)(l + 0) * FEA);
                       u0 = __builtin_nontemporal_load(em + (long)(l + 0) * FEA); }
        if (c1 >= 0) { w1 = __builtin_nontemporal_load(ew + (long)(l + 1) * FEA);
                       u1 = __builtin_nontemporal_load(em + (long)(l + 1) * FEA); }
        if (c2 >= 0) { w2 = __builtin_nontemporal_load(ew + (long)(l + 2) * FEA);
                       u2 = __builtin_nontemporal_load(em + (long)(l + 2) * FEA); }
        if (c3 >= 0) { w3 = __builtin_nontemporal_load(ew + (long)(l + 3) * FEA);
                       u3 = __builtin_nontemporal_load(em + (long)(l + 3) * FEA); }
        // ---- accumulate phase (fixed order) ----
        if (c0 >= 0) { acc[(c0 & 0xff) * 128 + tid]        += w0;
                       acc[((c0 >> 8) & 0xff) * 128 + tid] += u0; }
        if (c1 >= 0) { acc[(c1 & 0xff) * 128 + tid]        += w1;
                       acc[((c1 >> 8) & 0xff) * 128 + tid] += u1; }
        if (c2 >= 0) { acc[(c2 & 0xff) * 128 + tid]        += w2;
                       acc[((c2 >> 8) & 0xff) * 128 + tid] += u2; }
        if (c3 >= 0) { acc[(c3 & 0xff) * 128 + tid]        += w3;
                       acc[((c3 >> 8) & 0xff) * 128 + tid] += u3; }
    }
    __syncthreads();

    for (int a = 0; a < AA; ++a) {
        const float cnt = counts[b * AA + a];
        const float v = (cnt > 0.f) ? acc[a * 128 + tid] / fmaxf(cnt, 1.f) : 0.f;
        states[(long)b * SD + a * FEA + f0 + tid] = v;
    }
}

// ---------------------------------------------------------------------------
// Kernel 3: WMMA f32 GEMM  C[M,N] = act(A[M,K] @ W[K,N] + bias[N])
// One wave32 per 16x16 C tile, V_WMMA_F32_16X16X4_F32 along K, unroll 8.
// A frag (ISA 7.12.2, 16x4 f32): lanes 0-15 -> K=0,1 ; lanes 16-31 -> K=2,3.
// C frag: VGPR i -> M = i + 8*(lane>=16), N = lane&15.
// ACT: 0 = none, 1 = relu, 2 = abs.
// ---------------------------------------------------------------------------
template <int ACT>
__global__ void k_gemm_wmma(const float* __restrict__ Amat,   // [M,K]
                            const float* __restrict__ Wmat,   // [K,N]
                            const float* __restrict__ bias,   // [N]
                            float* __restrict__ Cmat,         // [M,N]
                            int M, int K, int N) {
    const int lane = threadIdx.x & 31;
    const int half = lane >> 4;
    const int r    = lane & 15;
    const int n0   = blockIdx.x * 16;
    const int m0   = blockIdx.y * 16;

    const float* arow = Amat + (long)(m0 + r) * K + 2 * half;  // 8B aligned (K even)
    const float* bcol = Wmat + n0 + r;

    v8f c = {};
#pragma unroll 8
    for (int k = 0; k < K; k += 4) {
        v2f a = *(const v2f*)(arow + k);                 // contiguous K pair
        v2f bfrag;
        bfrag.x = bcol[(long)(k + 2 * half + 0) * N];
        bfrag.y = bcol[(long)(k + 2 * half + 1) * N];
        c = __builtin_amdgcn_wmma_f32_16x16x4_f32(
                /*neg_a=*/false, a, /*neg_b=*/false, bfrag,
                /*c_mod=*/(short)0, c, /*reuse_a=*/false, /*reuse_b=*/false);
    }

    const float bv = bias[n0 + r];
#pragma unroll
    for (int i = 0; i < 8; ++i) {
        const int row = m0 + i + 8 * half;
        float v = c[i] + bv;
        if (ACT == 1) v = fmaxf(v, 0.f);
        if (ACT == 2) v = fabsf(v);
        Cmat[(long)row * N + n0 + r] = v;
    }
}

// ---------------------------------------------------------------------------
// Kernel 4: mixing tail. One block per b, 64 threads (one per embed dim).
// hidden = elu(agent_qs @ W1 + B1); q_tot = hidden . w_final + V1 . v2_w + v2_b
// ---------------------------------------------------------------------------
__global__ void k_final(const float* __restrict__ agent_qs,  // [B,A]
                        const float* __restrict__ W1,        // [B, A*E] (abs'd)
                        const float* __restrict__ B1,        // [B, E]
                        const float* __restrict__ WF,        // [B, E]   (abs'd)
                        const float* __restrict__ V1,        // [B, E]   (relu'd)
                        const float* __restrict__ v2_w,      // [E]
                        const float* __restrict__ v2_b,      // [1]
                        float* __restrict__ q_tot) {         // [B]
    const int b = blockIdx.x;
    const int e = threadIdx.x;     // 0..63
    __shared__ float red[EMB];

    float s = 0.f;
#pragma unroll
    for (int a = 0; a < AA; ++a)
        s += agent_qs[b * AA + a] * W1[(long)b * (AA * EMB) + a * EMB + e];
    s += B1[b * EMB + e];
    const float hidden = (s > 0.f) ? s : expm1f(s);   // elu, alpha=1

    red[e] = hidden * WF[b * EMB + e] + V1[b * EMB + e] * v2_w[e];
    __syncthreads();
    for (int off = 32; off > 0; off >>= 1) {
        if (e < off) red[e] += red[e + off];
        __syncthreads();
    }
    if (e == 0) q_tot[b] = red[0] + v2_b[0];
}

// ---------------------------------------------------------------------------
extern "C" void kernel_launch(void* const* d_in, const int* in_sizes, int n_in,
                              void* d_out, int out_size, void* d_ws, size_t ws_size,
                              hipStream_t stream) {
    const float*         qs_wt      = (const float*)d_in[0];
    const int*           aa_wt      = (const int*)d_in[1];
    const int*           aa_mut     = (const int*)d_in[2];
    const unsigned char* agent_mask = (const unsigned char*)d_in[3];
    const unsigned char* mask       = (const unsigned char*)d_in[4];
    const unsigned char* mut_mask   = (const unsigned char*)d_in[5];
    const float*         emb_wt     = (const float*)d_in[6];
    const float*         emb_mut    = (const float*)d_in[7];
    const float* w1_a = (const float*)d_in[8];
    const float* b1_a = (const float*)d_in[9];
    const float* w1_b = (const float*)d_in[10];
    const float* b1_b = (const float*)d_in[11];
    const float* hb_w = (const float*)d_in[12];
    const float* hb_b = (const float*)d_in[13];
    const float* wf_a = (const float*)d_in[14];
    const float* bf_a = (const float*)d_in[15];
    const float* wf_b = (const float*)d_in[16];
    const float* bf_b = (const float*)d_in[17];
    const float* v1_w = (const float*)d_in[18];
    const float* v1_b = (const float*)d_in[19];
    const float* v2_w = (const float*)d_in[20];
    const float* v2_b = (const float*)d_in[21];

    float* out    = (float*)d_out;
    float* q_tot  = out;            // [64]        (output #1)
    float* q_sele = out + BB;       // [64*1024]   (output #2)

    // workspace layout (floats)
    float* ws       = (float*)d_ws;
    float* states   = ws;                         // 64*10240 = 655360
    float* agent_qs = states   + (long)BB * SD;   // 1280
    float* counts   = agent_qs + BB * AA;         // 1280
    float* H1       = counts   + BB * AA;         // 64*512
    float* HF       = H1       + BB * HYP;        // 64*512
    float* V1       = HF       + BB * HYP;        // 64*64
    float* B1       = V1       + BB * EMB;        // 64*64
    float* W1       = B1       + BB * EMB;        // 64*1280
    float* WF       = W1       + BB * AA * EMB;   // 64*64

    // 1) gather + segment sums
    k_select<<<BB, 256, 0, stream>>>(qs_wt, aa_wt, aa_mut, agent_mask, mask,
                                     mut_mask, q_sele, agent_qs, counts);

    // 2) embedding segment-mean -> states  (HBM-bound streaming pass)
    k_gstate<<<BB * 4, 128, 0, stream>>>(aa_wt, aa_mut, agent_mask, mask,
                                         emb_wt, emb_mut, counts, states);

    // 3) hypernet GEMMs via v_wmma_f32_16x16x4_f32
    // H1 = relu(states @ w1_a + b1_a)   [64,512]
    k_gemm_wmma<1><<<dim3(HYP / 16, BB / 16), 32, 0, stream>>>(
        states, w1_a, b1_a, H1, BB, SD, HYP);
    // HF = relu(states @ wf_a + bf_a)   [64,512]
    k_gemm_wmma<1><<<dim3(HYP / 16, BB / 16), 32, 0, stream>>>(
        states, wf_a, bf_a, HF, BB, SD, HYP);
    // V1 = relu(states @ v1_w + v1_b)   [64,64]
    k_gemm_wmma<1><<<dim3(EMB / 16, BB / 16), 32, 0, stream>>>(
        states, v1_w, v1_b, V1, BB, SD, EMB);
    // B1 = states @ hb_w + hb_b         [64,64]
    k_gemm_wmma<0><<<dim3(EMB / 16, BB / 16), 32, 0, stream>>>(
        states, hb_w, hb_b, B1, BB, SD, EMB);
    // W1 = |H1 @ w1_b + b1_b|           [64,1280]
    k_gemm_wmma<2><<<dim3((AA * EMB) / 16, BB / 16), 32, 0, stream>>>(
        H1, w1_b, b1_b, W1, BB, HYP, AA * EMB);
    // WF = |HF @ wf_b + bf_b|           [64,64]
    k_gemm_wmma<2><<<dim3(EMB / 16, BB / 16), 32, 0, stream>>>(
        HF, wf_b, bf_b, WF, BB, HYP, EMB);

    // 4) mixing tail -> q_tot
    k_final<<<BB, EMB, 0, stream>>>(agent_qs, W1, B1, WF, V1, v2_w, v2_b, q_tot);
}